// CrossChannelInterp_72112500900725
// MI455X (gfx1250) — compile-verified
//
#include <hip/hip_runtime.h>

typedef __attribute__((ext_vector_type(2))) float v2f;
typedef __attribute__((ext_vector_type(4))) float v4f;
typedef __attribute__((ext_vector_type(8))) float v8f;

#define T_DIM 1024
#define D_DIM 128
#define ROW3D 384   // 3*D floats per (b,t) row
#define VSTR 132    // padded LDS row stride (dwords): 16B-aligned rows, conflict-free A reads

// ---------------- Pass 1: mean over T of the y-slice -> (B, D) ----------------
// 1024 threads: lane32 = tid&31 owns channels 4*lane32..+3 (b128 loads),
// chunk = tid>>5 (32 chunks x 32 t's). Fixed-order tree reduce -> deterministic.
__global__ __launch_bounds__(1024) void ccinterp_mean_kernel(
    const float* __restrict__ x, float* __restrict__ mean) {
  const int b = blockIdx.x;            // 0..127
  const int lane32 = threadIdx.x & 31;
  const int chunk = threadIdx.x >> 5;  // 0..31, each sums 32 t's
  __shared__ float part[32 * 128];     // 16 KB

  const float* px = x + ((size_t)b * T_DIM + (size_t)chunk * 32) * ROW3D + 4 * lane32;
  v4f s = {};
  for (int t = 0; t < 32; ++t) {
    v4f v = *(const v4f*)(px + (size_t)t * ROW3D);
    s += v;
  }
  *(v4f*)(part + chunk * 128 + 4 * lane32) = s;
  __syncthreads();
  if (threadIdx.x < 128) {
    float tot = 0.f;
#pragma unroll
    for (int c = 0; c < 32; ++c) tot += part[c * 128 + threadIdx.x];
    mean[b * D_DIM + threadIdx.x] = tot * (1.0f / (float)T_DIM);
  }
}

// ---------------- Pass 2: fused softmax + WMMA matmul + epilogue ----------------
// Block = 128 threads = 4 waves; each wave owns one 16-row (t) tile of one batch b.
__global__ __launch_bounds__(128) void ccinterp_fused_kernel(
    const float* __restrict__ x, const float* __restrict__ kern,
    const float* __restrict__ mean, float* __restrict__ out) {
  const int lane = threadIdx.x & 31;
  const int wave = threadIdx.x >> 5;       // 0..3
  const int l = lane & 15;                 // half-wave lane
  const int h = lane >> 4;                 // 0 or 1
  const int b = blockIdx.x >> 4;           // 16 blocks per batch
  const int tile = ((blockIdx.x & 15) << 2) + wave; // 0..63
  const int t0 = tile << 4;

  __shared__ float vbuf_all[4 * 16 * VSTR];        // 33 KB
  float* vbuf = vbuf_all + wave * (16 * VSTR);

  const float* meanb = mean + b * D_DIM;
  const v4f mean_l = *(const v4f*)(meanb + 4 * lane);  // lane owns channels 4*lane..+3

  const size_t rowbase0 = ((size_t)b * T_DIM + t0) * ROW3D;

  // ---- Phase 1: softmax over D, intensity, v = sm*(y-mean) into LDS ----
  for (int r = 0; r < 16; ++r) {
    const size_t base = rowbase0 + (size_t)r * ROW3D;
    const v4f yv = *(const v4f*)(x + base + 4 * lane);        // y slice (b128)
    const v4f wv = *(const v4f*)(x + base + 128 + 4 * lane);  // w slice (b128)

    float mx = fmaxf(fmaxf(wv.x, wv.y), fmaxf(wv.z, wv.w));
#pragma unroll
    for (int s = 16; s > 0; s >>= 1) mx = fmaxf(mx, __shfl_xor(mx, s, 32));

    v4f e;
    e.x = __expf(wv.x - mx); e.y = __expf(wv.y - mx);
    e.z = __expf(wv.z - mx); e.w = __expf(wv.w - mx);
    float sum = e.x + e.y + e.z + e.w;
#pragma unroll
    for (int s = 16; s > 0; s >>= 1) sum += __shfl_xor(sum, s, 32);

    const float inv = 1.0f / sum;
    const float emx = __expf(mx);  // exp(w) = e * exp(mx)

    *(v4f*)(out + base + 128 + 4 * lane) = e * emx;                  // intensity (b128)
    *(v4f*)(vbuf + r * VSTR + 4 * lane) = (e * inv) * (yv - mean_l); // v tile (b128)
  }

  // ---- Phase 2+3: smooth = v(16x128) @ kern(128x128) via V_WMMA_F32_16X16X4_F32 ----
  for (int nt = 0; nt < 8; ++nt) {
    const int ncol = nt * 16 + l;
    v8f acc = {};
#pragma unroll 4
    for (int ks = 0; ks < 32; ++ks) {
      const int k = ks * 4 + 2 * h;  // A: lanes0-15 K={0,1}, lanes16-31 K={2,3}
      v2f a = *(const v2f*)(vbuf + l * VSTR + k);
      v2f bb;
      bb.x = kern[k * D_DIM + ncol];            // B: same K-halving across lanes
      bb.y = kern[(k + 1) * D_DIM + ncol];
      acc = __builtin_amdgcn_wmma_f32_16x16x4_f32(
          /*neg_a=*/false, a, /*neg_b=*/false, bb,
          /*c_mod=*/(short)0, acc, /*reuse_a=*/false, /*reuse_b=*/false);
    }
    const float mn = meanb[ncol];
#pragma unroll
    for (int i = 0; i < 8; ++i) {               // C/D: VGPR i -> row i+8h, col ncol
      const int m = i + 8 * h;
      const size_t base = rowbase0 + (size_t)m * ROW3D;
      const float sm_v = acc[i] + mn;           // smooth
      const float yobs = x[base + 256 + ncol];  // y_obs slice
      out[base + ncol] = sm_v;                  // smooth
      out[base + 256 + ncol] = yobs - sm_v;     // y_trans
    }
  }
}

extern "C" void kernel_launch(void* const* d_in, const int* in_sizes, int n_in,
                              void* d_out, int out_size, void* d_ws, size_t ws_size,
                              hipStream_t stream) {
  (void)in_sizes; (void)n_in; (void)out_size; (void)ws_size;
  const float* x = (const float*)d_in[0];     // (128,1024,384) f32
  const float* kern = (const float*)d_in[1];  // (128,128) f32
  float* out = (float*)d_out;                 // (128,1024,384) f32
  float* mean = (float*)d_ws;                 // (128,128) f32 scratch

  ccinterp_mean_kernel<<<128, 1024, 0, stream>>>(x, mean);
  ccinterp_fused_kernel<<<2048, 128, 0, stream>>>(x, kern, mean, out);
}